// DepthConsistencyLoss_12266426597523
// MI455X (gfx1250) — compile-verified
//
#include <hip/hip_runtime.h>
#include <hip/hip_bf16.h>

typedef __attribute__((ext_vector_type(2))) float v2f;
typedef __attribute__((ext_vector_type(8))) float v8f;

#define IMG_H 224
#define IMG_W 224
#define NCH   21
#define NB    8
#define TILE  16
#define TS    18            // tile + 1-wide halo on each side
#define TSQ   (TS * TS)     // 324
#define NTHR  256

// spatial weights: exp(-dist2/50) for dist2 in {0,1,2}
#define WS_D0 1.0f
#define WS_D1 0.98019867f   // exp(-0.02)
#define WS_D2 0.96078944f   // exp(-0.04)

// ---- CDNA5 async global->LDS path (ASYNCcnt), with sync fallback ----
#if defined(__has_builtin)
#if __has_builtin(__builtin_amdgcn_global_load_async_to_lds_b32)
#define HAS_ASYNC_LDS 1
#endif
#endif
#ifndef HAS_ASYNC_LDS
#define HAS_ASYNC_LDS 0
#endif

typedef __attribute__((address_space(1))) int* glb_i32p;   // global (printed __device__ int*)
typedef __attribute__((address_space(3))) int* lds_i32p;   // LDS

__device__ __forceinline__ void wait_async_all() {
#if HAS_ASYNC_LDS
#if __has_builtin(__builtin_amdgcn_s_wait_asynccnt)
    __builtin_amdgcn_s_wait_asynccnt(0);
#else
    asm volatile("s_wait_asynccnt 0" ::: "memory");
#endif
#endif
}

// lane-XOR butterfly step via ds_swizzle (all lanes active): offset = xor<<10 | and_mask 0x1f
#define SWZ_XOR_ADD(s, m)                                                         \
    (s) += __int_as_float(__builtin_amdgcn_ds_swizzle(__float_as_int(s),          \
                                                      (((m) << 10) | 0x1f)))

__global__ void dcl_zero_kernel(float* out) { out[0] = 0.0f; }

// Stage an 18x18 zero-padded tile of one HxW channel into LDS.
// Async path: in-range elements via GLOBAL_LOAD_ASYNC_TO_LDS_B32 (EXEC-masked),
// out-of-range slots zero-filled with plain ds stores (DScnt).
__device__ __forceinline__ void stage_tile(float* __restrict__ dst,
                                           const float* __restrict__ src,
                                           int gx0, int gy0, int tid) {
    for (int i = tid; i < TSQ; i += NTHR) {
        int r = i / TS, cc = i - r * TS;
        int gy = gy0 + r, gx = gx0 + cc;
        if ((unsigned)gx < (unsigned)IMG_W && (unsigned)gy < (unsigned)IMG_H) {
#if HAS_ASYNC_LDS
            __builtin_amdgcn_global_load_async_to_lds_b32(
                (glb_i32p)(int*)const_cast<float*>(src + gy * IMG_W + gx),
                (lds_i32p)(int*)(dst + i),
                /*offset=*/0, /*cpol=*/0);
#else
            dst[i] = src[gy * IMG_W + gx];
#endif
        } else {
            dst[i] = 0.0f;
        }
    }
}

__global__ __launch_bounds__(NTHR)
void dcl_main_kernel(const float* __restrict__ cam,
                     const float* __restrict__ dep,
                     float* __restrict__ out) {
    __shared__ float sD[TSQ];
    __shared__ float sC9[TSQ], sC10[TSQ], sC11[TSQ];
    __shared__ float sBufA[TSQ], sBufB[TSQ];
    __shared__ float sCn[21 * NTHR];
    __shared__ float sRed;

    const int tid = threadIdx.x;
    const int lx = tid & 15, ly = tid >> 4;
    const int n  = blockIdx.z;
    const int gx0 = blockIdx.x * TILE - 1;
    const int gy0 = blockIdx.y * TILE - 1;

    const float* camN = cam + (size_t)n * NCH * IMG_H * IMG_W;
    const float* depN = dep + (size_t)n * IMG_H * IMG_W;

    if (tid == 0) sRed = 0.0f;

    // initial stage: depth + the three "center" channels + channel 0 into buffer A
    stage_tile(sD,   depN,                      gx0, gy0, tid);
    stage_tile(sC9,  camN +  9 * IMG_H * IMG_W, gx0, gy0, tid);
    stage_tile(sC10, camN + 10 * IMG_H * IMG_W, gx0, gy0, tid);
    stage_tile(sC11, camN + 11 * IMG_H * IMG_W, gx0, gy0, tid);
    stage_tile(sBufA, camN,                     gx0, gy0, tid);
    wait_async_all();
    __syncthreads();

    const int base = (ly + 1) * TS + (lx + 1);

    // per-pixel combined weights WW[t] = Ws[t] * exp(-(d_t - d_c)^2 * 50)
    const float ws[9] = { WS_D2, WS_D1, WS_D2,
                          WS_D1, WS_D0, WS_D1,
                          WS_D2, WS_D1, WS_D2 };
    float WW[9];
    {
        const float dcen = sD[base];
        #pragma unroll
        for (int t = 0; t < 9; ++t) {
            int off = (t / 3 - 1) * TS + (t % 3 - 1);
            float dd = sD[base + off] - dcen;
            WW[t] = ws[t] * __expf(-dd * dd * 50.0f);
        }
    }

    // "Center" vector Cn[j] = cam_unfold[84+j]: ch9 taps 3..8, ch10 taps 0..8, ch11 taps 0..5.
    // One LDS column per thread; row index j is wave-uniform -> conflict-free.
    #pragma unroll
    for (int t = 3; t <= 8; ++t) {
        int off = (t / 3 - 1) * TS + (t % 3 - 1);
        sCn[(t - 3) * NTHR + tid] = sC9[base + off];
    }
    #pragma unroll
    for (int t = 0; t <= 8; ++t) {
        int off = (t / 3 - 1) * TS + (t % 3 - 1);
        sCn[(6 + t) * NTHR + tid] = sC10[base + off];
    }
    #pragma unroll
    for (int t = 0; t <= 5; ++t) {
        int off = (t / 3 - 1) * TS + (t % 3 - 1);
        sCn[(15 + t) * NTHR + tid] = sC11[base + off];
    }
    // no barrier needed: each thread only reads its own sCn column

    // main loop: double-buffered channel streaming; async loads for channel c+1
    // are in flight while the 9 taps of channel c are computed.
    float acc = 0.0f;
    for (int c = 0; c < NCH; ++c) {
        const float* cur = (c & 1) ? sBufB : sBufA;
        float*       nxt = (c & 1) ? sBufA : sBufB;
        if (c + 1 < NCH)
            stage_tile(nxt, camN + (size_t)(c + 1) * IMG_H * IMG_W, gx0, gy0, tid);

        const int jbase = (9 * c) % 21;
        #pragma unroll
        for (int t = 0; t < 9; ++t) {
            int off = (t / 3 - 1) * TS + (t % 3 - 1);
            int j = jbase + t; if (j >= 21) j -= 21;   // (9c+t) mod 21
            float v = cur[base + off];
            float d = v - sCn[j * NTHR + tid];
            acc = fmaf(WW[t] * d, d, acc);
        }

        wait_async_all();        // next buffer resident
        __syncthreads();         // and all taps on cur finished before it is overwritten
    }

    // --- wave32 reduction via V_WMMA_F32_16X16X4_F32 ones-trick ---
    // A = ones(16x4); B carries the 32 per-lane partials (b0) and zeros (b1).
    // D[0][n] = sum_k B[k][n]; summing D row 0 over 16 columns = wave total.
    v2f a; a[0] = 1.0f; a[1] = 1.0f;
    v2f b; b[0] = acc;  b[1] = 0.0f;
    v8f cz = {0.0f, 0.0f, 0.0f, 0.0f, 0.0f, 0.0f, 0.0f, 0.0f};
    v8f dm = __builtin_amdgcn_wmma_f32_16x16x4_f32(
        /*neg_a=*/false, a, /*neg_b=*/false, b,
        /*c_mod=*/(short)0, cz, /*reuse_a=*/false, /*reuse_b=*/false);
    float s = dm[0];             // lanes 0-15: D[0][n]; lanes 16-31: identical row D[8][n]
    SWZ_XOR_ADD(s, 8);
    SWZ_XOR_ADD(s, 4);
    SWZ_XOR_ADD(s, 2);
    SWZ_XOR_ADD(s, 1);           // lane 0 (and 16) now hold the full wave sum

    if ((tid & 31) == 0) atomicAdd(&sRed, s);
    __syncthreads();
    if (tid == 0)
        atomicAdd(out, sRed * (1.0f / (float)(NB * IMG_H * IMG_W)));
}

extern "C" void kernel_launch(void* const* d_in, const int* in_sizes, int n_in,
                              void* d_out, int out_size, void* d_ws, size_t ws_size,
                              hipStream_t stream) {
    const float* cam = (const float*)d_in[0];   // (8, 21, 224, 224) f32
    const float* dep = (const float*)d_in[1];   // (8, 1, 224, 224) f32
    float* out = (float*)d_out;                 // scalar f32

    dcl_zero_kernel<<<1, 1, 0, stream>>>(out);

    dim3 grid(IMG_W / TILE, IMG_H / TILE, NB);  // 14 x 14 x 8
    dcl_main_kernel<<<grid, NTHR, 0, stream>>>(cam, dep, out);
}